// PACNet_4810363372849
// MI455X (gfx1250) — compile-verified
//
#include <hip/hip_runtime.h>
#include <hip/hip_bf16.h>

// ---------------- WMMA types (wave32, gfx1250) ----------------
typedef float v2f __attribute__((ext_vector_type(2)));
typedef float v8f __attribute__((ext_vector_type(8)));

#define HEADS 4
#define DH 16
#define INDIM 300
#define HID 64
#define OUT2 32
#define SLOPE 0.2f

// ------------- helpers -------------
__device__ __forceinline__ unsigned int f2ord(float f) {
    unsigned int u = __float_as_uint(f);
    return (u & 0x80000000u) ? ~u : (u | 0x80000000u);
}
__device__ __forceinline__ float ord2f(unsigned int u) {
    return (u & 0x80000000u) ? __uint_as_float(u & 0x7FFFFFFFu)
                             : __uint_as_float(~u);
}
__device__ __forceinline__ float lrelu(float e) { return e > 0.f ? e : SLOPE * e; }

// ------------- init kernels -------------
__global__ void k_zero_u(unsigned int* __restrict__ p, long long total) {
    long long t = (long long)blockIdx.x * blockDim.x + threadIdx.x;
    if (t < total) p[t] = 0u;
}
__global__ void k_bcast(float* __restrict__ p, const float* __restrict__ v,
                        int mask, long long total) {
    long long t = (long long)blockIdx.x * blockDim.x + threadIdx.x;
    if (t < total) p[t] = v[t & mask];
}

// ------------- GEMM1: h[N,64] = x[N,300] @ W1[300,64]  (fp32 WMMA 16x16x4) ----
__global__ __launch_bounds__(128) void k_gemm1(const float* __restrict__ x,
                                               const float* __restrict__ W1,
                                               float* __restrict__ h, int n) {
    const int lane = threadIdx.x & 31;
    const int wave = threadIdx.x >> 5;
    const int l  = lane & 15;     // column within tile / row for A
    const int hi = lane >> 4;     // K half: lanes 0-15 -> K0,1 ; lanes 16-31 -> K2,3
    const int rw = blockIdx.x * 64 + wave * 16;

    v8f acc[4] = {};
    int arow = rw + l;
    if (arow >= n) arow = n - 1;              // clamp: EXEC must stay all-ones
    const float* ap = x + (size_t)arow * INDIM;

    for (int kk = 0; kk < INDIM; kk += 4) {
        const int k0 = kk + hi * 2;
        float2 av = *(const float2*)(ap + k0);     // row stride 1200B -> 8B aligned
        v2f a; a.x = av.x; a.y = av.y;
        const float* bp = W1 + (size_t)k0 * HID + l;
#pragma unroll
        for (int c = 0; c < 4; ++c) {
            v2f b; b.x = bp[c * 16]; b.y = bp[c * 16 + HID];
            acc[c] = __builtin_amdgcn_wmma_f32_16x16x4_f32(
                false, a, false, b, (short)0, acc[c], false, false);
        }
    }
    // epilogue: uniform fast path for full tiles (only last block is partial)
    float* hp = h + (size_t)(rw + hi * 8) * HID + l;
    if (rw + 16 <= n) {
#pragma unroll
        for (int c = 0; c < 4; ++c)
#pragma unroll
            for (int r = 0; r < 8; ++r)
                hp[(size_t)r * HID + c * 16] = acc[c][r];
    } else {
#pragma unroll
        for (int c = 0; c < 4; ++c)
#pragma unroll
            for (int r = 0; r < 8; ++r) {
                int row = rw + hi * 8 + r;    // C layout: lanes16-31 hold M=8..15
                if (row < n) h[(size_t)row * HID + c * 16 + l] = acc[c][r];
            }
    }
}

// ------------- GEMM2: h2[N,32] = relu(out1)[N,64] @ W2[64,32] -------------
__global__ __launch_bounds__(128) void k_gemm2(const float* __restrict__ x1pre,
                                               const float* __restrict__ W2,
                                               float* __restrict__ h2, int n) {
    const int lane = threadIdx.x & 31;
    const int wave = threadIdx.x >> 5;
    const int l  = lane & 15;
    const int hi = lane >> 4;
    const int rw = blockIdx.x * 64 + wave * 16;

    v8f acc[2] = {};
    int arow = rw + l;
    if (arow >= n) arow = n - 1;
    const float* ap = x1pre + (size_t)arow * HID;

    for (int kk = 0; kk < HID; kk += 4) {
        const int k0 = kk + hi * 2;
        float2 av = *(const float2*)(ap + k0);
        v2f a; a.x = fmaxf(av.x, 0.f); a.y = fmaxf(av.y, 0.f);   // fused ReLU
        const float* bp = W2 + (size_t)k0 * OUT2 + l;
#pragma unroll
        for (int c = 0; c < 2; ++c) {
            v2f b; b.x = bp[c * 16]; b.y = bp[c * 16 + OUT2];
            acc[c] = __builtin_amdgcn_wmma_f32_16x16x4_f32(
                false, a, false, b, (short)0, acc[c], false, false);
        }
    }
    float* op = h2 + (size_t)(rw + hi * 8) * OUT2 + l;
    if (rw + 16 <= n) {
#pragma unroll
        for (int c = 0; c < 2; ++c)
#pragma unroll
            for (int r = 0; r < 8; ++r)
                op[(size_t)r * OUT2 + c * 16] = acc[c][r];
    } else {
#pragma unroll
        for (int c = 0; c < 2; ++c)
#pragma unroll
            for (int r = 0; r < 8; ++r) {
                int row = rw + hi * 8 + r;
                if (row < n) h2[(size_t)row * OUT2 + c * 16 + l] = acc[c][r];
            }
    }
}

// ------------- attention scalars: a_s/a_d [N,4] -------------
__global__ void k_attn(const float* __restrict__ h,
                       const float* __restrict__ att_src,
                       const float* __restrict__ att_dst,
                       float* __restrict__ a_s, float* __restrict__ a_d, int n) {
    int t = blockIdx.x * blockDim.x + threadIdx.x;
    if (t >= n * HEADS) return;
    int node = t >> 2, hh = t & 3;
    const float* hp = h + (size_t)node * HID + hh * DH;
    const float* as = att_src + hh * DH;
    const float* ad = att_dst + hh * DH;
    float ss = 0.f, sd = 0.f;
#pragma unroll
    for (int d = 0; d < DH; ++d) { float v = hp[d]; ss += v * as[d]; sd += v * ad[d]; }
    a_s[t] = ss; a_d[t] = sd;
}

// ------------- GAT segment max (ordered-uint atomicMax) -------------
__global__ void k_edge_max(const int* __restrict__ ei, long long E, int n,
                           const float* __restrict__ a_s, const float* __restrict__ a_d,
                           unsigned int* __restrict__ m_u) {
    long long t = (long long)blockIdx.x * blockDim.x + threadIdx.x;
    if (t >= E + n) return;
    int src, dst;
    if (t < E) { src = ei[t]; dst = ei[E + t]; } else { src = dst = (int)(t - E); }
#pragma unroll
    for (int hh = 0; hh < HEADS; ++hh) {
        float e = lrelu(a_s[src * 4 + hh] + a_d[dst * 4 + hh]);
        atomicMax(&m_u[dst * 4 + hh], f2ord(e));
    }
}

// ------------- GAT softmax denominator -------------
__global__ void k_edge_denom(const int* __restrict__ ei, long long E, int n,
                             const float* __restrict__ a_s, const float* __restrict__ a_d,
                             const unsigned int* __restrict__ m_u,
                             float* __restrict__ denom) {
    long long t = (long long)blockIdx.x * blockDim.x + threadIdx.x;
    if (t >= E + n) return;
    int src, dst;
    if (t < E) { src = ei[t]; dst = ei[E + t]; } else { src = dst = (int)(t - E); }
#pragma unroll
    for (int hh = 0; hh < HEADS; ++hh) {
        float e = lrelu(a_s[src * 4 + hh] + a_d[dst * 4 + hh]);
        float m = ord2f(m_u[dst * 4 + hh]);
        atomicAdd(&denom[dst * 4 + hh], expf(e - m));
    }
}

// ------------- GAT aggregation: out1[dst] += alpha * h[src]  -------------
// one thread per (edge, lane<32); each thread handles features lane and lane+32
__global__ void k_edge_agg(const int* __restrict__ ei, long long E, int n,
                           const float* __restrict__ a_s, const float* __restrict__ a_d,
                           const unsigned int* __restrict__ m_u,
                           const float* __restrict__ denom,
                           const float* __restrict__ h, float* __restrict__ out1) {
    long long t = (long long)blockIdx.x * blockDim.x + threadIdx.x;
    long long total = (E + n) * 32;
    if (t >= total) return;
    long long e = t >> 5;
    int lane = (int)(t & 31);
    int src, dst;
    if (e < E) { src = ei[e]; dst = ei[E + e]; } else { src = dst = (int)(e - E); }
    int hb = lane >> 4;
#pragma unroll
    for (int half = 0; half < 2; ++half) {
        int hh = half * 2 + hb;
        int j  = lane + half * 32;
        float ev = lrelu(a_s[src * 4 + hh] + a_d[dst * 4 + hh]);
        float m  = ord2f(m_u[dst * 4 + hh]);
        float alpha = expf(ev - m) / denom[dst * 4 + hh];
        atomicAdd(&out1[(size_t)dst * HID + j], h[(size_t)src * HID + j] * alpha);
    }
}

// ------------- GCN degree -------------
__global__ void k_deg(const int* __restrict__ ei, long long E, int n,
                      float* __restrict__ deg) {
    long long t = (long long)blockIdx.x * blockDim.x + threadIdx.x;
    if (t >= E + n) return;
    int dst = (t < E) ? ei[E + t] : (int)(t - E);
    atomicAdd(&deg[dst], 1.0f);
}
__global__ void k_dinv(float* __restrict__ deg, int n) {
    int t = blockIdx.x * blockDim.x + threadIdx.x;
    if (t >= n) return;
    float d = deg[t];
    deg[t] = (d > 0.f) ? rsqrtf(d) : 0.f;
}

// ------------- GCN aggregation: out[dst] += norm * h2[src] -------------
__global__ void k_gcn_agg(const int* __restrict__ ei, long long E, int n,
                          const float* __restrict__ dinv,
                          const float* __restrict__ h2, float* __restrict__ out) {
    long long t = (long long)blockIdx.x * blockDim.x + threadIdx.x;
    long long total = (E + n) * 32;
    if (t >= total) return;
    long long e = t >> 5;
    int j = (int)(t & 31);
    int src, dst;
    if (e < E) { src = ei[e]; dst = ei[E + e]; } else { src = dst = (int)(e - E); }
    float norm = dinv[src] * dinv[dst];
    atomicAdd(&out[(size_t)dst * OUT2 + j], h2[(size_t)src * OUT2 + j] * norm);
}

// ------------- host launch -------------
static inline int nblk(long long total, int bs) { return (int)((total + bs - 1) / bs); }

extern "C" void kernel_launch(void* const* d_in, const int* in_sizes, int n_in,
                              void* d_out, int out_size, void* d_ws, size_t ws_size,
                              hipStream_t stream) {
    const float* x       = (const float*)d_in[0];
    const float* W1      = (const float*)d_in[1];
    const float* att_src = (const float*)d_in[2];
    const float* att_dst = (const float*)d_in[3];
    const float* b1      = (const float*)d_in[4];
    const float* W2      = (const float*)d_in[5];
    const float* b2      = (const float*)d_in[6];
    const int*   ei      = (const int*)d_in[7];
    const int*   ei2     = (const int*)d_in[8];

    const int n        = in_sizes[0] / INDIM;        // 100000
    const long long E1 = in_sizes[7] / 2;            // 3.2M
    const long long E2 = in_sizes[8] / 2;

    // workspace layout (fp32 unless noted), everything L2-resident (192MB L2)
    float*        h     = (float*)d_ws;                       // [N,64]
    float*        out1  = h    + (size_t)n * HID;             // [N,64] (init b1)
    float*        h2    = out1 + (size_t)n * HID;             // [N,32]
    float*        a_s   = h2   + (size_t)n * OUT2;            // [N,4]
    float*        a_d   = a_s  + (size_t)n * HEADS;           // [N,4]
    unsigned int* m_u   = (unsigned int*)(a_d + (size_t)n * HEADS); // [N,4] ordered-uint
    float*        denom = (float*)(m_u + (size_t)n * HEADS);  // [N,4]
    float*        deg   = denom + (size_t)n * HEADS;          // [N] -> dinv in place

    const int BS = 256;

    // zero m_u | denom | deg (contiguous: n*9 dwords)
    k_zero_u<<<nblk((long long)n * 9, BS), BS, 0, stream>>>(m_u, (long long)n * 9);
    // out1 <- b1 (bias folded into accumulator init)
    k_bcast<<<nblk((long long)n * HID, BS), BS, 0, stream>>>(out1, b1, HID - 1, (long long)n * HID);
    // out <- b2
    k_bcast<<<nblk((long long)n * OUT2, BS), BS, 0, stream>>>((float*)d_out, b2, OUT2 - 1, (long long)n * OUT2);

    // GAT layer
    k_gemm1<<<(n + 63) / 64, 128, 0, stream>>>(x, W1, h, n);
    k_attn<<<nblk((long long)n * HEADS, BS), BS, 0, stream>>>(h, att_src, att_dst, a_s, a_d, n);
    k_edge_max<<<nblk(E1 + n, BS), BS, 0, stream>>>(ei, E1, n, a_s, a_d, m_u);
    k_edge_denom<<<nblk(E1 + n, BS), BS, 0, stream>>>(ei, E1, n, a_s, a_d, m_u, denom);
    k_edge_agg<<<nblk((E1 + n) * 32, BS), BS, 0, stream>>>(ei, E1, n, a_s, a_d, m_u, denom, h, out1);

    // GCN layer (ReLU + x1@W2 fused in k_gemm2)
    k_gemm2<<<(n + 63) / 64, 128, 0, stream>>>(out1, W2, h2, n);
    k_deg<<<nblk(E2 + n, BS), BS, 0, stream>>>(ei2, E2, n, deg);
    k_dinv<<<nblk(n, BS), BS, 0, stream>>>(deg, n);
    k_gcn_agg<<<nblk((E2 + n) * 32, BS), BS, 0, stream>>>(ei2, E2, n, deg, h2, (float*)d_out);
}